// GATLayer_2688649527647
// MI455X (gfx1250) — compile-verified
//
#include <hip/hip_runtime.h>
#include <math.h>

// GAT forward, MI455X (gfx1250), wave32, FP32 WMMA (16x16x4).
// B=32, C=N=64, T=256, D=64, H=4, F=32, HF=128, E=512 (+64 self loops), G=B*T=8192.

typedef float v2f __attribute__((ext_vector_type(2)));
typedef float v8f __attribute__((ext_vector_type(8)));

#define XS 68   // LDS row stride for X / attention matrix A (16B-aligned rows, conflict-free)
#define WS 66   // LDS row stride for transposed W_aug / transposed h_aug (8B-aligned, conflict-free)

__device__ __forceinline__ float leaky(float v) { return v > 0.f ? v : 0.2f * v; }

// CAS-based float max on LDS (guaranteed-compiling; lowers to ds_cmpstore loop)
__device__ __forceinline__ void atomicMaxF(float* addr, float val) {
  int* ia = (int*)addr;
  int old = __float_as_int(*addr);
  while (__int_as_float(old) < val) {
    int assumed = old;
    old = atomicCAS(ia, assumed, __float_as_int(val));
    if (old == assumed) break;
  }
}

__global__ __launch_bounds__(256)
void gat_fused_kernel(const float* __restrict__ x,      // [32,64,256,64]
                      const int*   __restrict__ eidx,   // [2,512]
                      const float* __restrict__ W,      // [64,128]
                      const float* __restrict__ a_src,  // [4,32]
                      const float* __restrict__ a_dst,  // [4,32]
                      const float* __restrict__ bias,   // [128]
                      float* __restrict__ out)          // [32,64,256,128]
{
  __shared__ float s_X[64 * XS];     // X tile; reused later as dense attention matrix A
  __shared__ float s_Wt[144 * WS];   // W_aug transposed: s_Wt[col*WS + k]
  __shared__ float s_ht[144 * WS];   // h_aug transposed: s_ht[col*WS + node]
  __shared__ float s_e[576 * 4];     // per-(edge,head): e -> exp -> alpha
  __shared__ float s_m[64 * 4];      // per-(dst,head) running max
  __shared__ float s_den[64 * 4];    // per-(dst,head) softmax denom
  __shared__ int   s_src[576];
  __shared__ int   s_dst[576];
  __shared__ float s_bias[128];

  const int tid  = threadIdx.x;
  const int lane = tid & 31;
  const int w    = tid >> 5;          // wave id, 0..7
  const int g    = blockIdx.x;        // graph id = b*T + t
  const int b    = g >> 8;            // T = 256
  const int t    = g & 255;

  // ---------------- stage: X (coalesced float4), W_aug^T, edges, bias ----------------
  {
    const int row = tid >> 2;              // 0..63  (node)
    const int c0  = (tid & 3) << 4;        // 0,16,32,48
    const float* src = x + (((size_t)(b * 64 + row) * 256 + t) * 64 + c0);
    float* dst = &s_X[row * XS + c0];
#pragma unroll
    for (int q = 0; q < 4; ++q)
      *(float4*)(dst + 4 * q) = *(const float4*)(src + 4 * q);
  }
  // W_aug columns: 0..127 = W, 128..131 = W@a_src per head, 132..135 = W@a_dst, 136..143 = 0
  for (int idx = tid; idx < 144 * 64; idx += 256) {
    const int j = idx >> 6;   // augmented column 0..143
    const int d = idx & 63;   // input feature
    float v;
    if (j < 128) {
      v = W[d * 128 + j];
    } else if (j < 136) {
      const int hh = (j - 128) & 3;
      const float* av = (j < 132) ? a_src : a_dst;
      float s = 0.f;
#pragma unroll
      for (int f = 0; f < 32; ++f) s = fmaf(W[d * 128 + hh * 32 + f], av[hh * 32 + f], s);
      v = s;
    } else {
      v = 0.f;
    }
    s_Wt[j * WS + d] = v;
  }
  for (int i = tid; i < 512; i += 256) { s_src[i] = eidx[i]; s_dst[i] = eidx[512 + i]; }
  if (tid < 64)  { s_src[512 + tid] = tid; s_dst[512 + tid] = tid; }   // self loops
  if (tid < 128) s_bias[tid] = bias[tid];
  s_m[tid]   = -INFINITY;   // 256 entries, covered exactly
  s_den[tid] = 0.f;
  __syncthreads();

  // ---------------- GEMM1: h_aug(64x144) = X(64x64) @ W_aug(64x144) via WMMA ----------------
  {
    const int m0   = (w & 3) << 4;
    const int arow = m0 + (lane & 15);
    const int koff = (lane >> 4) << 1;          // 0 or 2 (K pair select per lane half)
    v2f afrag[16];
    const float* xr = &s_X[arow * XS + koff];
#pragma unroll
    for (int kk = 0; kk < 16; ++kk) afrag[kk] = *(const v2f*)(xr + 4 * kk);

    for (int ct = (w >> 2); ct < 9; ct += 2) {  // waves 0-3: cols {0,2,4,6,8}; 4-7: {1,3,5,7}
      const int n0  = ct << 4;
      const int col = n0 + (lane & 15);
      const float* wc = &s_Wt[col * WS + koff];
      v8f c = {};
#pragma unroll
      for (int kk = 0; kk < 16; ++kk) {
        v2f bfrag = *(const v2f*)(wc + 4 * kk);
        c = __builtin_amdgcn_wmma_f32_16x16x4_f32(false, afrag[kk], false, bfrag,
                                                  (short)0, c, false, false);
      }
      // store D tile transposed into s_ht: lane (row m0+v / m0+8+v, col n0+lane&15)
      const int rbase = m0 + ((lane >> 4) << 3);
      float* hcol = &s_ht[col * WS + rbase];
#pragma unroll
      for (int v = 0; v < 8; ++v) hcol[v] = c[v];
    }
  }
  __syncthreads();

  // alpha_src / alpha_dst live in transposed-h columns 128..135
  const float* s_as = &s_ht[128 * WS];   // s_as[hh*WS + node]
  const float* s_ad = &s_ht[132 * WS];

  // ---------------- edge pass 1: e = leaky(as[src]+ad[dst]); segment max ----------------
  for (int idx = tid; idx < 576 * 4; idx += 256) {
    const int i  = idx >> 2;
    const int hh = idx & 3;
    const int sN = s_src[i], dN = s_dst[i];
    const float e = leaky(s_as[hh * WS + sN] + s_ad[hh * WS + dN]);
    s_e[idx] = e;
    atomicMaxF(&s_m[dN * 4 + hh], e);
  }
  __syncthreads();
  // ---------------- edge pass 2: exp(e - m[dst]); segment sum ----------------
  for (int idx = tid; idx < 576 * 4; idx += 256) {
    const int i  = idx >> 2;
    const int hh = idx & 3;
    const int dN = s_dst[i];
    const float ex = __expf(s_e[idx] - s_m[dN * 4 + hh]);
    s_e[idx] = ex;
    atomicAdd(&s_den[dN * 4 + hh], ex);
  }
  __syncthreads();
  // ---------------- edge pass 3: alpha = ex / denom[dst] ----------------
  for (int idx = tid; idx < 576 * 4; idx += 256) {
    const int i  = idx >> 2;
    const int hh = idx & 3;
    s_e[idx] = s_e[idx] / s_den[s_dst[i] * 4 + hh];
  }
  __syncthreads();

  // ---------------- per head: dense A[dst,src] then out = A @ h_head (WMMA) ----------------
  float* s_A = s_X;                         // X no longer needed; reuse 64x68 buffer
  const int m0    = (w & 3) << 4;           // output row tile
  const int n0    = (w >> 2) << 4;          // output col tile within 32-wide head (0 or 16)
  const int koff  = (lane >> 4) << 1;
  const int arow  = m0 + (lane & 15);
  const int rbase = m0 + ((lane >> 4) << 3);

  for (int hh = 0; hh < 4; ++hh) {
    for (int idx = tid; idx < 64 * 64; idx += 256)
      s_A[(idx >> 6) * XS + (idx & 63)] = 0.f;
    __syncthreads();
    for (int i = tid; i < 576; i += 256)
      atomicAdd(&s_A[s_dst[i] * XS + s_src[i]], s_e[i * 4 + hh]);
    __syncthreads();

    const int ch = hh * 32 + n0 + (lane & 15);      // output channel for this lane column
    const float* Ar = &s_A[arow * XS + koff];
    const float* hc = &s_ht[ch * WS + koff];
    v8f c = {};
#pragma unroll
    for (int kk = 0; kk < 16; ++kk) {
      v2f af = *(const v2f*)(Ar + 4 * kk);
      v2f bf = *(const v2f*)(hc + 4 * kk);
      c = __builtin_amdgcn_wmma_f32_16x16x4_f32(false, af, false, bf, (short)0, c, false, false);
    }
    // fused epilogue: + bias, leaky-relu, store to out[b, node, t, ch]
    const float bv = s_bias[ch];
    float* op = out + (((size_t)(b * 64 + rbase) * 256 + t) * 128 + ch);
#pragma unroll
    for (int v = 0; v < 8; ++v)
      op[(size_t)v * 256 * 128] = leaky(c[v] + bv);
    __syncthreads();   // protect s_A re-zero next head
  }
}

extern "C" void kernel_launch(void* const* d_in, const int* in_sizes, int n_in,
                              void* d_out, int out_size, void* d_ws, size_t ws_size,
                              hipStream_t stream) {
  (void)in_sizes; (void)n_in; (void)out_size; (void)d_ws; (void)ws_size;
  const float* x    = (const float*)d_in[0];
  const int*   eidx = (const int*)d_in[1];
  const float* W    = (const float*)d_in[2];
  const float* a_s  = (const float*)d_in[3];
  const float* a_d  = (const float*)d_in[4];
  const float* bias = (const float*)d_in[5];
  float* out = (float*)d_out;

  dim3 grid(32 * 256);   // G = B*T = 8192 graphs, one workgroup each
  dim3 block(256);       // 8 waves (wave32)
  gat_fused_kernel<<<grid, block, 0, stream>>>(x, eidx, W, a_s, a_d, bias, out);
}